// NopadBaichuanAttention_85633057947690
// MI455X (gfx1250) — compile-verified
//
#include <hip/hip_runtime.h>
#include <hip/hip_bf16.h>

// Problem constants (from reference)
#define B_   4
#define S_   1024
#define H_   32
#define D_   128
#define BS_  16
#define NB_  64
#define E_   4096   // H*D
#define T_   4096   // B*S

typedef float v2f __attribute__((ext_vector_type(2)));
typedef float v8f __attribute__((ext_vector_type(8)));

// CDNA5 fp32 WMMA: D(16x16 f32) = A(16x4 f32) * B(4x16 f32) + C
__device__ __forceinline__ v8f wmma4(v2f a, v2f b, v8f c) {
  return __builtin_amdgcn_wmma_f32_16x16x4_f32(false, a, false, b, (short)0, c,
                                               false, false);
}

// ---------------------------------------------------------------------------
// Tiled fp32-WMMA GEMM: C[M x N] = A[M x K] * B[K x N], all row-major.
// Block = 256 threads (8 waves) computing a 256x128 tile; each wave owns a
// 64x64 sub-tile (16 v8f accumulators).  K-step 16, double-buffered LDS with
// register staging so global loads overlap the 64 WMMAs per stage.
// isqkv==1: grid.z selects qkv_weight slice; z=0 -> plain store to Cq,
//           z=1/2 -> scatter into k/v paged cache layout via block_tables.
// ---------------------------------------------------------------------------
__global__ __launch_bounds__(256) void gemm256x128_wmma(
    const float* __restrict__ A, const float* __restrict__ Bmat,
    float* __restrict__ Cq, float* __restrict__ kcache,
    float* __restrict__ vcache, const int* __restrict__ btab,
    int K, int N, int isqkv)
{
  __shared__ float sA[2][256 * 16];   // 2 x 16 KB
  __shared__ float sB[2][16 * 128];   // 2 x 8 KB

  const int tid  = threadIdx.x;
  const int z    = blockIdx.z;
  const int bm   = blockIdx.y * 256;
  const int bn   = blockIdx.x * 128;
  const float* Bp = Bmat + (size_t)z * K * N;

  const int w    = tid >> 5;        // wave 0..7
  const int lane = tid & 31;
  const int m16  = lane & 15;       // M (A frag) / N (B frag) index
  const int hh   = lane >> 4;       // half-wave selects K pair 0,1 vs 2,3
  const int wm   = (w & 3) * 64;    // 4 waves along M
  const int wn   = (w >> 2) * 64;   // 2 waves along N

  v8f acc[4][4];
  #pragma unroll
  for (int tm = 0; tm < 4; tm++)
    #pragma unroll
    for (int tn = 0; tn < 4; tn++) acc[tm][tn] = (v8f){0,0,0,0,0,0,0,0};

  // Per-thread staging registers: A tile 256x16 = 1024 f4 (4/thread),
  // B tile 16x128 = 512 f4 (2/thread).
  float4 ra[4], rb[2];

  #define FETCH(K0)                                                          \
    {                                                                        \
      _Pragma("unroll")                                                      \
      for (int t = 0; t < 4; t++) {                                          \
        int q = tid + t * 256;                                               \
        ra[t] = *(const float4*)&A[(size_t)(bm + (q >> 2)) * K + (K0) +      \
                                   ((q & 3) << 2)];                          \
      }                                                                      \
      _Pragma("unroll")                                                      \
      for (int t = 0; t < 2; t++) {                                          \
        int q = tid + t * 256;                                               \
        rb[t] = *(const float4*)&Bp[(size_t)((K0) + (q >> 5)) * N + bn +     \
                                    ((q & 31) << 2)];                        \
      }                                                                      \
    }

  #define STORE_STAGE(BUF)                                                   \
    {                                                                        \
      _Pragma("unroll")                                                      \
      for (int t = 0; t < 4; t++)                                            \
        *(float4*)&sA[BUF][(tid + t * 256) << 2] = ra[t];                    \
      _Pragma("unroll")                                                      \
      for (int t = 0; t < 2; t++)                                            \
        *(float4*)&sB[BUF][(tid + t * 256) << 2] = rb[t];                    \
    }

  // Prologue: fill stage 0
  FETCH(0)
  STORE_STAGE(0)

  int cur = 0;
  for (int k0 = 0; k0 < K; k0 += 16) {
    __syncthreads();  // stage `cur` visible; prev compute on `cur^1` done
    const bool next = (k0 + 16) < K;
    if (next) {
      FETCH(k0 + 16)                     // global loads in flight over compute
      if (k0 + 32 < K) {                 // warm L2 one more stage ahead
        __builtin_prefetch(&A[(size_t)(bm + (tid >> 2)) * K + k0 + 32], 0, 1);
        __builtin_prefetch(&Bp[(size_t)(k0 + 32 + (tid >> 5)) * N + bn], 0, 1);
      }
    }

    #pragma unroll
    for (int kk = 0; kk < 16; kk += 4) {
      v2f a[4], b[4];
      #pragma unroll
      for (int tm = 0; tm < 4; tm++)     // A frag: ds_load_b64
        a[tm] = *(const v2f*)&sA[cur][(wm + tm * 16 + m16) * 16 + kk + 2 * hh];
      #pragma unroll
      for (int tn = 0; tn < 4; tn++) {   // B frag: row-of-K striped over lanes
        b[tn].x = sB[cur][(kk + 2 * hh)     * 128 + wn + tn * 16 + m16];
        b[tn].y = sB[cur][(kk + 2 * hh + 1) * 128 + wn + tn * 16 + m16];
      }
      #pragma unroll
      for (int tm = 0; tm < 4; tm++)
        #pragma unroll
        for (int tn = 0; tn < 4; tn++)
          acc[tm][tn] = wmma4(a[tm], b[tn], acc[tm][tn]);
    }

    if (next) STORE_STAGE(cur ^ 1)
    cur ^= 1;
  }
  #undef FETCH
  #undef STORE_STAGE

  // Epilogue.  C/D layout: VGPR r holds M = r + 8*hh, N = m16.
  #pragma unroll
  for (int tm = 0; tm < 4; tm++) {
    #pragma unroll
    for (int tn = 0; tn < 4; tn++) {
      #pragma unroll
      for (int r = 0; r < 8; r++) {
        int row = bm + wm + tm * 16 + r + 8 * hh;
        int col = bn + wn + tn * 16 + m16;
        float val = acc[tm][tn][r];
        if (isqkv && z != 0) {
          // scatter into paged cache: (blk, h, bs, d)
          int b  = row >> 10, s = row & 1023, nb = s >> 4, bs = s & 15;
          int blk = btab[b * NB_ + nb];
          int h  = col >> 7, d = col & 127;
          float* dst = (z == 1) ? kcache : vcache;
          dst[(size_t)(((blk * H_ + h) << 4) + bs) * 128 + d] = val;
        } else {
          Cq[(size_t)row * N + col] = val;
        }
      }
    }
  }
}

// ---------------------------------------------------------------------------
// RoPE, in place.  y=0: Q in workspace (T,E) layout.  y=1: K in cache layout.
// ---------------------------------------------------------------------------
__global__ __launch_bounds__(256) void rope_kernel(
    float* __restrict__ q, float* __restrict__ kc,
    const float* __restrict__ cosb, const float* __restrict__ sinb,
    const int* __restrict__ btab)
{
  unsigned idx = blockIdx.x * 256u + threadIdx.x;  // 0 .. T*H*64
  int i = idx & 63;
  int h = (idx >> 6) & 31;
  int t = idx >> 11;
  float c  = cosb[t * 64 + i];
  float sn = sinb[t * 64 + i];
  if (blockIdx.y == 0) {
    size_t base = (size_t)t * E_ + h * 128;
    float x1 = q[base + i], x2 = q[base + 64 + i];
    q[base + i]      = x1 * c - x2 * sn;
    q[base + 64 + i] = x2 * c + x1 * sn;
  } else {
    int b = t >> 10, s = t & 1023, nb = s >> 4, bs = s & 15;
    int blk = btab[b * NB_ + nb];
    size_t base = (size_t)(((blk * H_ + h) << 4) + bs) * 128;
    float x1 = kc[base + i], x2 = kc[base + 64 + i];
    kc[base + i]      = x1 * c - x2 * sn;
    kc[base + 64 + i] = x2 * c + x1 * sn;
  }
}

// ---------------------------------------------------------------------------
// Flash-style causal attention, one (64 q-rows, head) tile per block.
// 128 threads = 4 waves; each wave owns 16 q-rows.  Q·K^T and P·V via
// v_wmma_f32_16x16x4_f32; online softmax with 16-lane shuffle reductions.
// K/V come from the paged caches (one key block = one contiguous 16x128
// tile).  Output overwrites the Q workspace.
// ---------------------------------------------------------------------------
__global__ __launch_bounds__(128) void attn_kernel(
    const float* __restrict__ kc, const float* __restrict__ vc,
    float* __restrict__ qws, const int* __restrict__ btab)
{
  __shared__ float sQ[64 * 128];   // 32 KB, pre-scaled by 1/sqrt(D)
  __shared__ float sK[16 * 128];   // 8 KB
  __shared__ float sV[16 * 128];   // 8 KB
  __shared__ float sP[4 * 256];    // 4 KB, per-wave P transpose staging

  const int tid   = threadIdx.x;
  const int bh    = blockIdx.y;
  const int b     = bh >> 5;       // H = 32
  const int h     = bh & 31;
  const int qbase = blockIdx.x * 64;
  const int w     = tid >> 5;      // wave 0..3
  const int lane  = tid & 31;
  const int nn    = lane & 15;
  const int hh    = lane >> 4;
  const float scale = 0.08838834764831845f;  // 1/sqrt(128)

  for (int i = tid; i < 64 * 32; i += 128) {
    int r = i >> 5, c4 = (i & 31) << 2;
    float4 v = *(const float4*)&qws[(size_t)(b * S_ + qbase + r) * E_ + h * 128 + c4];
    v.x *= scale; v.y *= scale; v.z *= scale; v.w *= scale;
    *(float4*)&sQ[r * 128 + c4] = v;
  }

  const int qb_my = (qbase >> 4) + w;  // this wave's diagonal key block
  const int jend  = (qbase >> 4) + 3;

  float m_i[8], l_i[8];
  v8f o[8];
  #pragma unroll
  for (int r = 0; r < 8; r++) { m_i[r] = -__builtin_inff(); l_i[r] = 0.f; }
  #pragma unroll
  for (int t = 0; t < 8; t++) o[t] = (v8f){0,0,0,0,0,0,0,0};

  float* sPw = &sP[w * 256];

  for (int j = 0; j <= jend; j++) {
    __syncthreads();
    {   // stage K/V block j (contiguous 16x128 cache tiles)
      int blk = btab[b * NB_ + j];
      const float* kg = &kc[(size_t)(blk * H_ + h) * (BS_ * D_)];
      const float* vg = &vc[(size_t)(blk * H_ + h) * (BS_ * D_)];
      for (int i = tid; i < 512; i += 128) {
        *(float4*)&sK[i << 2] = *(const float4*)&kg[i << 2];
        *(float4*)&sV[i << 2] = *(const float4*)&vg[i << 2];
      }
    }
    __syncthreads();
    if (j > qb_my) continue;  // wave-uniform causal skip

    // S = Q(16x128) * K^T : 32 WMMA K-steps.  B-frag of K^T == A-layout of K.
    v8f s = (v8f){0,0,0,0,0,0,0,0};
    #pragma unroll
    for (int k0 = 0; k0 < 128; k0 += 4) {
      v2f a  = *(const v2f*)&sQ[(w * 16 + nn) * 128 + k0 + 2 * hh];
      v2f bb = *(const v2f*)&sK[nn * 128 + k0 + 2 * hh];
      s = wmma4(a, bb, s);
    }

    if (j == qb_my) {  // diagonal masking: key > query row
      #pragma unroll
      for (int r = 0; r < 8; r++) {
        int qrow = qbase + w * 16 + r + 8 * hh;
        if (j * 16 + nn > qrow) s[r] = -__builtin_inff();
      }
    }

    // Online softmax: a score row lives across 16 lanes within one VGPR
    float alpha[8], p[8];
    #pragma unroll
    for (int r = 0; r < 8; r++) {
      float mx = s[r];
      #pragma unroll
      for (int off = 8; off > 0; off >>= 1) mx = fmaxf(mx, __shfl_xor(mx, off, 16));
      float mnew = fmaxf(m_i[r], mx);
      p[r] = __expf(s[r] - mnew);
      float su = p[r];
      #pragma unroll
      for (int off = 8; off > 0; off >>= 1) su += __shfl_xor(su, off, 16);
      alpha[r] = __expf(m_i[r] - mnew);
      l_i[r]   = l_i[r] * alpha[r] + su;
      m_i[r]   = mnew;
    }
    #pragma unroll
    for (int t = 0; t < 8; t++)
      #pragma unroll
      for (int r = 0; r < 8; r++) o[t][r] *= alpha[r];

    // Transpose P from C/D layout to A layout via per-wave LDS staging
    #pragma unroll
    for (int r = 0; r < 8; r++) sPw[(r + 8 * hh) * 16 + nn] = p[r];
    asm volatile("s_wait_dscnt 0" ::: "memory");

    // O += P(16x16) * V(16x128)
    #pragma unroll
    for (int k0 = 0; k0 < 16; k0 += 4) {
      v2f a = *(const v2f*)&sPw[nn * 16 + k0 + 2 * hh];
      #pragma unroll
      for (int t = 0; t < 8; t++) {
        v2f bb;
        bb.x = sV[(k0 + 2 * hh)     * 128 + t * 16 + nn];
        bb.y = sV[(k0 + 2 * hh + 1) * 128 + t * 16 + nn];
        o[t] = wmma4(a, bb, o[t]);
      }
    }
  }

  // Normalize and overwrite the Q workspace slot (same elements this block read)
  #pragma unroll
  for (int t = 0; t < 8; t++) {
    #pragma unroll
    for (int r = 0; r < 8; r++) {
      int qrow = qbase + w * 16 + r + 8 * hh;
      qws[(size_t)(b * S_ + qrow) * E_ + h * 128 + t * 16 + nn] = o[t][r] / l_i[r];
    }
  }
}

// ---------------------------------------------------------------------------
extern "C" void kernel_launch(void* const* d_in, const int* in_sizes, int n_in,
                              void* d_out, int out_size, void* d_ws, size_t ws_size,
                              hipStream_t stream) {
  const float* hidden = (const float*)d_in[0];
  const float* qkvw   = (const float*)d_in[1];
  const float* ow     = (const float*)d_in[2];
  const float* cosb   = (const float*)d_in[3];
  const float* sinb   = (const float*)d_in[4];
  const int*   btab   = (const int*)d_in[7];

  float* out    = (float*)d_out;                 // (T, E)
  float* kcache = out + (size_t)T_ * E_;         // (B*NB, H, BS, D)
  float* vcache = out + 2 * (size_t)T_ * E_;
  float* wsq    = (float*)d_ws;                  // T*E fp32 (64 MB)

  dim3 blk256(256), blk128(128);

  // 1) QKV projection: Q -> ws, K/V -> paged caches (scatter in epilogue)
  gemm256x128_wmma<<<dim3(E_ / 128, T_ / 256, 3), blk256, 0, stream>>>(
      hidden, qkvw, wsq, kcache, vcache, btab, E_, E_, 1);

  // 2) RoPE on Q (ws) and K (cache), in place
  rope_kernel<<<dim3((unsigned)(T_ * H_ * 64 / 256), 2), blk256, 0, stream>>>(
      wsq, kcache, cosb, sinb, btab);

  // 3) Causal flash attention; output overwrites Q workspace
  attn_kernel<<<dim3(S_ / 64, B_ * H_), blk128, 0, stream>>>(
      kcache, vcache, wsq, btab);

  // 4) Output projection into d_out[0:T*E]
  gemm256x128_wmma<<<dim3(E_ / 128, T_ / 256, 1), blk256, 0, stream>>>(
      wsq, ow, out, nullptr, nullptr, nullptr, E_, E_, 0);
}